// ReservoirLinearRNN_Block_81114752352326
// MI455X (gfx1250) — compile-verified
//
#include <hip/hip_runtime.h>
#include <hip/hip_bf16.h>

typedef __attribute__((ext_vector_type(2))) float v2f;
typedef __attribute__((ext_vector_type(8))) float v8f;

#define B_   16
#define S_   4096
#define D_   256
#define H_   256
#define MH_  1024
#define EPS_ 1e-5f

// D = A(16x4) * B(4x16) + C(16x16), full f32 via the CDNA5 matrix core.
static __device__ __forceinline__ v8f wmma_f32(v2f a, v2f b, v8f c) {
  return __builtin_amdgcn_wmma_f32_16x16x4_f32(false, a, false, b, (short)0, c,
                                               false, false);
}

// ---------------------------------------------------------------------------
// Kernel 1: Bu[bs][h] = sum_d x[bs][d] * B_w[h][d]   (written into hs region)
// One wave per 16x16 output tile; K=256 swept in steps of 4.
// ---------------------------------------------------------------------------
__global__ __launch_bounds__(256) void bu_gemm_kernel(
    const float* __restrict__ x, const float* __restrict__ Bw,
    float* __restrict__ bu) {
  const int lane  = threadIdx.x & 31;
  const int wave  = threadIdx.x >> 5;
  const int gw    = blockIdx.x * 8 + wave;   // 0..65535 wave tiles
  const int mt    = gw >> 4;                 // 0..4095 row tiles
  const int nt    = gw & 15;                 // 0..15   col tiles
  const int l15   = lane & 15;
  const int koff  = (lane >> 4) * 2;
  const int mbase = (lane >> 4) * 8;

  const float* xrow = x  + (size_t)(mt * 16 + l15) * D_;  // A-frag row (M=l15)
  const float* brow = Bw + (size_t)(nt * 16 + l15) * D_;  // B-frag: B_w[n][k]

  v8f acc = {};
#pragma unroll 8
  for (int k0 = 0; k0 < D_; k0 += 4) {
    v2f a = *(const v2f*)(xrow + k0 + koff);
    v2f b = *(const v2f*)(brow + k0 + koff);
    acc = wmma_f32(a, b, acc);
  }
  float* orow = bu + (size_t)(mt * 16) * H_ + nt * 16 + l15;
#pragma unroll
  for (int j = 0; j < 8; ++j)
    orow[(size_t)(mbase + j) * H_] = acc[j];
}

// ---------------------------------------------------------------------------
// Kernel 2: sequential scan h_t = h_{t-1} @ A + Bu_t, in place over hs.
// Single workgroup, 16 waves. Each wave holds its 256x16 slice of A as 64
// register-resident B-fragments; h ping-pongs through LDS (stride 260 to
// avoid bank conflicts). One barrier per step.
// ---------------------------------------------------------------------------
__global__ __launch_bounds__(512) void scan_kernel(
    const float* __restrict__ A, float* __restrict__ hs) {
  extern __shared__ float hbuf[];            // 2 buffers of 16 x 260
  const int HP    = 260;
  const int lane  = threadIdx.x & 31;
  const int wave  = threadIdx.x >> 5;        // N-tile owner, 0..15
  const int l15   = lane & 15;
  const int koff  = (lane >> 4) * 2;
  const int mbase = (lane >> 4) * 8;
  const int ng    = wave * 16 + l15;

  // Preload this wave's column slice of A: 64 K-fragments in VGPRs.
  v2f bfrag[64];
#pragma unroll
  for (int i = 0; i < 64; ++i) {
    const int k = 4 * i + koff;
    v2f t;
    t.x = A[(size_t)k * H_ + ng];
    t.y = A[(size_t)(k + 1) * H_ + ng];
    bfrag[i] = t;
  }

  for (int i = threadIdx.x; i < 2 * 16 * HP; i += blockDim.x) hbuf[i] = 0.f;
  __syncthreads();

  int cur = 0;
#pragma unroll 1
  for (int t = 0; t < S_; ++t) {
    // Accumulator seeded with Bu[:, t, tile] (in-place region).
    v8f acc;
#pragma unroll
    for (int j = 0; j < 8; ++j)
      acc[j] = hs[((size_t)(mbase + j) * S_ + t) * H_ + ng];
    if (t + 1 < S_)  // hint the next step's Bu tile toward the caches
      __builtin_prefetch(&hs[((size_t)mbase * S_ + (t + 1)) * H_ + ng], 0, 0);

    const float* hrow = hbuf + cur * 16 * HP + l15 * HP + koff;
#pragma unroll
    for (int i = 0; i < 64; ++i) {
      v2f a = *(const v2f*)(hrow + 4 * i);   // ds_load_b64, conflict-free
      acc = wmma_f32(a, bfrag[i], acc);
    }

    float* dst = hbuf + (cur ^ 1) * 16 * HP;
#pragma unroll
    for (int j = 0; j < 8; ++j) {
      const int m = mbase + j;
      dst[m * HP + ng] = acc[j];                              // next-step h
      hs[((size_t)m * S_ + t) * H_ + ng] = acc[j];            // H_seq_pre out
    }
    cur ^= 1;
    __syncthreads();
  }
}

// ---------------------------------------------------------------------------
// Kernel 3: LN(z_last) -> MLP.  Only t = S-1 of H_seq is ever consumed.
// One workgroup, 16 waves: per-batch LN, then two WMMA GEMMs via LDS.
// ---------------------------------------------------------------------------
__global__ __launch_bounds__(512) void head_kernel(
    const float* __restrict__ x, const float* __restrict__ hs,
    const float* __restrict__ ln_g, const float* __restrict__ ln_b,
    const float* __restrict__ w1, const float* __restrict__ b1,
    const float* __restrict__ w2, const float* __restrict__ b2,
    float* __restrict__ out) {
  extern __shared__ float smem[];
  const int HP = 260, AP = 1028;             // padded strides (stride%64==4)
  float* hl  = smem;                         // 16 x 260   normalized h_last
  float* act = smem + 16 * HP;               // 16 x 1028  relu activations
  const int lane = threadIdx.x & 31;
  const int wave = threadIdx.x >> 5;

  {  // LayerNorm: one wave per batch row, 8 elements per lane.
    const int b = wave;
    const float* hp = hs + ((size_t)b * S_ + (S_ - 1)) * H_;
    const float* xp = x  + ((size_t)b * S_ + (S_ - 1)) * D_;
    float z[8], s = 0.f;
#pragma unroll
    for (int i = 0; i < 8; ++i) {
      const int h = lane * 8 + i;
      z[i] = hp[h] + xp[h];
      s += z[i];
    }
#pragma unroll
    for (int off = 16; off >= 1; off >>= 1) s += __shfl_xor(s, off, 32);
    const float mu = s * (1.f / H_);
    float v = 0.f;
#pragma unroll
    for (int i = 0; i < 8; ++i) { const float d = z[i] - mu; v += d * d; }
#pragma unroll
    for (int off = 16; off >= 1; off >>= 1) v += __shfl_xor(v, off, 32);
    const float rstd = rsqrtf(v * (1.f / H_) + EPS_);
#pragma unroll
    for (int i = 0; i < 8; ++i) {
      const int h = lane * 8 + i;
      hl[b * HP + h] = (z[i] - mu) * rstd * ln_g[h] + ln_b[h];
    }
  }
  __syncthreads();

  const int l15   = lane & 15;
  const int koff  = (lane >> 4) * 2;
  const int mbase = (lane >> 4) * 8;

  // MLP layer 1: act = relu(hl @ w1^T + b1); 64 N-tiles, 4 per wave.
#pragma unroll 1
  for (int tile = 0; tile < 4; ++tile) {
    const int ng = (wave * 4 + tile) * 16 + l15;
    const float* wrow = w1 + (size_t)ng * H_;
    const float* hrow = hl + l15 * HP + koff;
    v8f acc = {};
#pragma unroll
    for (int k0 = 0; k0 < H_; k0 += 4) {
      v2f a = *(const v2f*)(hrow + k0);
      v2f b = *(const v2f*)(wrow + k0 + koff);
      acc = wmma_f32(a, b, acc);
    }
    const float bias = b1[ng];
#pragma unroll
    for (int j = 0; j < 8; ++j)
      act[(mbase + j) * AP + ng] = fmaxf(acc[j] + bias, 0.f);
  }
  __syncthreads();

  // MLP layer 2: out = act @ w2^T + b2; 16 N-tiles, one per wave, K=1024.
  {
    const int ng = wave * 16 + l15;
    const float* wrow = w2 + (size_t)ng * MH_;
    const float* arow = act + l15 * AP + koff;
    v8f acc = {};
#pragma unroll 8
    for (int k0 = 0; k0 < MH_; k0 += 4) {
      v2f a = *(const v2f*)(arow + k0);
      v2f b = *(const v2f*)(wrow + k0 + koff);
      acc = wmma_f32(a, b, acc);
    }
    const float bias = b2[ng];
#pragma unroll
    for (int j = 0; j < 8; ++j)
      out[(mbase + j) * H_ + ng] = acc[j] + bias;
  }
}

// ---------------------------------------------------------------------------
extern "C" void kernel_launch(void* const* d_in, const int* in_sizes, int n_in,
                              void* d_out, int out_size, void* d_ws,
                              size_t ws_size, hipStream_t stream) {
  const float* x    = (const float*)d_in[0];
  const float* A    = (const float*)d_in[1];
  const float* Bw   = (const float*)d_in[2];
  const float* ln_g = (const float*)d_in[3];
  const float* ln_b = (const float*)d_in[4];
  const float* w1   = (const float*)d_in[5];
  const float* b1   = (const float*)d_in[6];
  const float* w2   = (const float*)d_in[7];
  const float* b2   = (const float*)d_in[8];

  float* out = (float*)d_out;                 // [16*256]   MLP output
  float* hs  = out + (size_t)B_ * H_;         // [16*4096*256] H_seq_pre

  // 1) Bu written straight into the H_seq_pre region (consumed in place).
  bu_gemm_kernel<<<8192, 256, 0, stream>>>(x, Bw, hs);
  // 2) Sequential recurrence: one persistent workgroup, A in VGPRs.
  scan_kernel<<<1, 512, 2 * 16 * 260 * sizeof(float), stream>>>(A, hs);
  // 3) LN(last) + MLP head.
  head_kernel<<<1, 512, (16 * 260 + 16 * 1028) * sizeof(float), stream>>>(
      x, hs, ln_g, ln_b, w1, b1, w2, b2, out);
}